// Model_74938589381132
// MI455X (gfx1250) — compile-verified
//
#include <hip/hip_runtime.h>
#include <math.h>

typedef __attribute__((ext_vector_type(2))) float v2f;
typedef __attribute__((ext_vector_type(4))) float v4f;
typedef __attribute__((ext_vector_type(8))) float v8f;

#define B_  4096
#define T_  256
#define I_  64
#define H_  64
#define ROWS 16          // batch rows per block (one WMMA M-tile)
#define LSTR 68          // LDS row stride in floats: 68 ≡ 4 (mod 64) -> conflict-free b64 frag loads

__global__ __launch_bounds__(128) void rnn_fused_f32_wmma(
    const float* __restrict__ x,      // [B,T,I]
    const float* __restrict__ h0,     // [1,B,H]
    const float* __restrict__ W_ih,   // [H,I]
    const float* __restrict__ W_hh,   // [H,H]
    const float* __restrict__ b_ih,   // [H]
    const float* __restrict__ b_hh,   // [H]
    float* __restrict__ out_hT,       // [B,H]   (first region of d_out)
    float* __restrict__ out)          // [B,T,H] (second region of d_out)
{
    __shared__ __align__(16) float xs[2][ROWS * LSTR];  // x_t tile, double buffered
    __shared__ __align__(16) float hs[2][ROWS * LSTR];  // h tile,   double buffered

    const int tid     = threadIdx.x;
    const int lane    = tid & 31;
    const int wave    = tid >> 5;        // 0..3 -> N-tile of 16 columns
    const int rowbase = blockIdx.x * ROWS;

    const int nloc  = lane & 15;         // N within tile
    const int khalf = lane >> 4;         // K-group select (lanes 16-31 start at K+2)
    const int n     = wave * 16 + nloc;  // output column 0..63

    // ---- B fragments (weights) -> registers, loaded once, reused for all T steps ----
    v2f bx[16], bh[16];
#pragma unroll
    for (int j = 0; j < 16; ++j) {
        const int k = 4 * j + 2 * khalf;               // VGPR0 <- K=k, VGPR1 <- K=k+1
        bx[j] = *(const v2f*)(W_ih + (size_t)n * I_ + k);
        bh[j] = *(const v2f*)(W_hh + (size_t)n * H_ + k);
    }
    const float bias = b_ih[n] + b_hh[n];

    // ---- prologue: cooperative load of x_0 and h_0 tiles into LDS buffers 0 ----
#pragma unroll
    for (int it = 0; it < 2; ++it) {
        const int f  = tid + it * 128;                 // float4 index: 16 rows x 16 float4
        const int r  = f >> 4, c4 = f & 15;
        v4f vx = *(const v4f*)(x  + ((size_t)(rowbase + r) * T_ + 0) * I_ + c4 * 4);
        v4f vh = *(const v4f*)(h0 +  (size_t)(rowbase + r) * H_       + c4 * 4);
        *(v4f*)(&xs[0][r * LSTR + c4 * 4]) = vx;
        *(v4f*)(&hs[0][r * LSTR + c4 * 4]) = vh;
    }
    __syncthreads();

    // per-lane A-fragment base offset (floats) and D-fragment row base
    const int aoff  = nloc * LSTR + 2 * khalf;  // A-matrix M == lane&15 (same as nloc)
    const int mbase = 8 * khalf;

    for (int t = 0; t < T_; ++t) {
        const int  p    = t & 1;
        const bool more = (t + 1) < T_;

        // prefetch x_{t+1} into registers (loads issue before the WMMA chain)
        v4f pf0 = {}, pf1 = {};
        if (more) {
            const int f0 = tid,       r0 = f0 >> 4, c0 = f0 & 15;
            const int f1 = tid + 128, r1 = f1 >> 4, c1 = f1 & 15;
            pf0 = *(const v4f*)(x + ((size_t)(rowbase + r0) * T_ + (t + 1)) * I_ + c0 * 4);
            pf1 = *(const v4f*)(x + ((size_t)(rowbase + r1) * T_ + (t + 1)) * I_ + c1 * 4);
        }

        // ---- two independent WMMA chains (halved serial depth):
        //      accX = X_t[16x64] * Wih_tile,  accH = H[16x64] * Whh_tile ----
        v8f accX = {};
        v8f accH = {};
        const float* xrow = &xs[p][aoff];
        const float* hrow = &hs[p][aoff];
#pragma unroll
        for (int j = 0; j < 16; ++j) {
            v2f ax = *(const v2f*)(xrow + 4 * j);
            accX = __builtin_amdgcn_wmma_f32_16x16x4_f32(false, ax, false, bx[j],
                                                         (short)0, accX, false, false);
            v2f ah = *(const v2f*)(hrow + 4 * j);
            accH = __builtin_amdgcn_wmma_f32_16x16x4_f32(false, ah, false, bh[j],
                                                         (short)0, accH, false, false);
        }

        // ---- epilogue: combine chains + bias + tanh; write h + out; hT at the end ----
        float* hdst = &hs[1 - p][0];
#pragma unroll
        for (int r = 0; r < 8; ++r) {
            const int   m = r + mbase;                 // D-frag: lanes 16-31 hold rows +8
            const float v = tanhf(accX[r] + accH[r] + bias);
            hdst[m * LSTR + n] = v;
            out[((size_t)(rowbase + m) * T_ + t) * H_ + n] = v;
            if (t == T_ - 1)
                out_hT[(size_t)(rowbase + m) * H_ + n] = v;
        }

        // spill prefetched x_{t+1} into the alternate buffer
        if (more) {
            const int f0 = tid,       r0 = f0 >> 4, c0 = f0 & 15;
            const int f1 = tid + 128, r1 = f1 >> 4, c1 = f1 & 15;
            *(v4f*)(&xs[1 - p][r0 * LSTR + c0 * 4]) = pf0;
            *(v4f*)(&xs[1 - p][r1 * LSTR + c1 * 4]) = pf1;
        }
        __syncthreads();   // publishes h_{t+1} and x_{t+1} tiles
    }
}

extern "C" void kernel_launch(void* const* d_in, const int* in_sizes, int n_in,
                              void* d_out, int out_size, void* d_ws, size_t ws_size,
                              hipStream_t stream) {
    const float* x    = (const float*)d_in[0];
    const float* h0   = (const float*)d_in[1];   // [1,B,H]
    const float* W_ih = (const float*)d_in[2];
    const float* W_hh = (const float*)d_in[3];
    const float* b_ih = (const float*)d_in[4];
    const float* b_hh = (const float*)d_in[5];

    float* out_hT = (float*)d_out;               // first tuple element: [1,B,H]
    float* out    = (float*)d_out + (size_t)B_ * H_;  // second: [B*T, H]

    dim3 grid(B_ / ROWS);   // 256 blocks
    dim3 block(128);        // 4 waves: one 16-column N-tile each
    rnn_fused_f32_wmma<<<grid, block, 0, stream>>>(x, h0, W_ih, W_hh, b_ih, b_hh,
                                                   out_hT, out);
}